// InteractionEncoder_65412351918170
// MI455X (gfx1250) — compile-verified
//
#include <hip/hip_runtime.h>
#include <hip/hip_bf16.h>
#include <math.h>

typedef float v2f __attribute__((ext_vector_type(2)));
typedef float v8f __attribute__((ext_vector_type(8)));

#define MAXD_INV (1.0f / 50.0f)

// ---------------------------------------------------------------------------
// Kernel 1: precompute PA[b,n,:] = slots[b,n] @ W1[0:256]   + b1   (half 0)
//                       PB[b,n,:] = slots[b,n] @ W1[256:512]        (half 1)
// fp32 WMMA 16x16x4, one wave per 16x16 output tile, K = 256.
// slots viewed as [2048, 256] row-major; ws viewed as [2][2048][256].
// ---------------------------------------------------------------------------
__global__ __launch_bounds__(32) void ie_k1_precompute(
    const float* __restrict__ slots,  // [2048,256]
    const float* __restrict__ W1,     // [516,256]
    const float* __restrict__ b1,     // [256]
    float* __restrict__ ws)           // [2][2048][256]
{
    const int lane = threadIdx.x & 31;
    const int rt   = blockIdx.x;      // row tile (0..127)
    const int ct   = blockIdx.y;      // col tile (0..15)
    const int half = blockIdx.z;      // 0 -> PA rows of W1, 1 -> PB rows

    const int mlo = lane & 15;        // A row / B,C column within tile
    const int khi = (lane >> 4) << 1; // 0 for lanes 0-15, 2 for lanes 16-31

    const int rowA = rt * 16 + mlo;
    const int col  = ct * 16 + mlo;

    const float* arow  = slots + (size_t)rowA * 256;
    const float* bcol  = W1 + (size_t)(half * 256) * 256 + col;

    v8f c = {};
#pragma unroll 4
    for (int k0 = 0; k0 < 256; k0 += 4) {
        const int ka = k0 + khi;
        // A fragment: (M = lane%16, K = ka, ka+1)
        v2f a = *(const v2f*)(arow + ka);
        // B fragment: (K = ka, ka+1, N = col)
        v2f bm;
        bm.x = bcol[(size_t)ka * 256];
        bm.y = bcol[(size_t)(ka + 1) * 256];
        c = __builtin_amdgcn_wmma_f32_16x16x4_f32(false, a, false, bm,
                                                  (short)0, c, false, false);
    }

    const float bias = (half == 0) ? b1[col] : 0.0f;
    float* out = ws + (size_t)half * 2048 * 256;
    const int rbase = rt * 16 + ((lane >> 4) << 3);  // C layout: rows r / r+8
#pragma unroll
    for (int r = 0; r < 8; ++r) {
        out[(size_t)(rbase + r) * 256 + col] = c[r] + bias;
    }
}

// ---------------------------------------------------------------------------
// Kernel 2: per (b, j-block of 16, i): fuse extra-feature MLP input, relu,
// GEMM2 (h @ W2) via fp32 WMMA, bias+mask, feats store, itypes & causal heads.
// Block = 256 threads = 8 waves, each wave owns one i; W2 staged via LDS in
// two 64KB k-halves.
// ---------------------------------------------------------------------------
__global__ __launch_bounds__(256) void ie_k2_main(
    const float* __restrict__ positions, // [2048,4]
    const float* __restrict__ em,        // [2048]
    const float* __restrict__ W1,        // [516,256] (rows 512..515 used)
    const float* __restrict__ W2,        // [256,128]
    const float* __restrict__ b2,        // [128]
    const float* __restrict__ Wt,        // [128,8]
    const float* __restrict__ bt,        // [8]
    const float* __restrict__ Wc,        // [128,1]
    const float* __restrict__ bc,        // [1]
    const float* __restrict__ pa,        // [2048,256]
    const float* __restrict__ pb,        // [2048,256]
    float* __restrict__ feats_out,       // [131072,128]
    float* __restrict__ itypes_out,      // [131072,8]
    float* __restrict__ causal_out)      // [131072]
{
    __shared__ float w2s[128 * 128];     // 64KB: half of W2 ([k 0..127][d])

    const int tid  = threadIdx.x;
    const int lane = tid & 31;
    const int wave = tid >> 5;
    const int b    = blockIdx.x;             // 0..31
    const int j0   = blockIdx.y * 16;        // 0,16,32,48
    const int i    = blockIdx.z * 8 + wave;  // 0..63

    const int mA  = lane & 15;               // A-layout pair index
    const int khi = (lane >> 4) << 1;        // 0 or 2
    const int hi  = lane >> 4;               // 0/1
    const int nc  = lane & 15;               // C-layout column within tile

    // Geometric features for pair (i, j0+mA)
    const float* pi = positions + (size_t)(b * 64 + i) * 4;
    const float* pj = positions + (size_t)(b * 64 + j0 + mA) * 4;
    const float rx = pi[0] - pj[0];
    const float ry = pi[1] - pj[1];
    const float d2 = rx * rx + ry * ry;
    const float dist = (d2 == 0.0f) ? 0.0f : sqrtf(d2);
    const float dn  = dist * MAXD_INV;
    const float dnc = fminf(dn, 1.0f);

    const float* paRow = pa + (size_t)(b * 64 + i) * 256;
    const float* pbRow = pb + (size_t)(b * 64 + j0 + mA) * 256;
    const float* w1c   = W1 + (size_t)512 * 256;

    v8f c[8];
#pragma unroll
    for (int dt = 0; dt < 8; ++dt) c[dt] = v8f{};

    const float4* w2g4 = (const float4*)W2;
    float4* w2s4 = (float4*)w2s;

    for (int half = 0; half < 2; ++half) {
        __syncthreads();
        // Stage 64KB of W2 (k rows half*128 .. half*128+127)
#pragma unroll
        for (int it = 0; it < 16; ++it)
            w2s4[it * 256 + tid] = w2g4[half * 4096 + it * 256 + tid];
        __syncthreads();

        const int kbeg = half * 128;
#pragma unroll 2
        for (int k0 = kbeg; k0 < kbeg + 128; k0 += 4) {
            const int ka = k0 + khi;
            // Build A fragment: relu(PA + PB + e . W1c) at (mA, ka / ka+1)
            v2f va = *(const v2f*)(paRow + ka);
            v2f vb = *(const v2f*)(pbRow + ka);
            v2f c0 = *(const v2f*)(w1c + ka);
            v2f c1 = *(const v2f*)(w1c + 256 + ka);
            v2f c2 = *(const v2f*)(w1c + 512 + ka);
            v2f c3 = *(const v2f*)(w1c + 768 + ka);
            v2f a;
            a.x = fmaxf(fmaf(dnc, c3.x, fmaf(dn, c2.x,
                        fmaf(ry, c1.x, fmaf(rx, c0.x, va.x + vb.x)))), 0.0f);
            a.y = fmaxf(fmaf(dnc, c3.y, fmaf(dn, c2.y,
                        fmaf(ry, c1.y, fmaf(rx, c0.y, va.y + vb.y)))), 0.0f);
            const int kl = ka - kbeg;
#pragma unroll
            for (int dt = 0; dt < 8; ++dt) {
                v2f bm;
                bm.x = w2s[kl * 128 + dt * 16 + mA];
                bm.y = w2s[(kl + 1) * 128 + dt * 16 + mA];
                c[dt] = __builtin_amdgcn_wmma_f32_16x16x4_f32(
                    false, a, false, bm, (short)0, c[dt], false, false);
            }
        }
    }

    // ---- Epilogue: bias + mask, store feats ----
    const float em_i = em[b * 64 + i];
    float biasv[8];
#pragma unroll
    for (int dt = 0; dt < 8; ++dt) biasv[dt] = b2[dt * 16 + nc];
    float maskr[8];
#pragma unroll
    for (int r = 0; r < 8; ++r)
        maskr[r] = em_i * em[b * 64 + j0 + r + 8 * hi];

    const size_t pairBase = (size_t)(b * 64 + i) * 64 + j0;
#pragma unroll
    for (int dt = 0; dt < 8; ++dt) {
#pragma unroll
        for (int r = 0; r < 8; ++r) {
            float f = (c[dt][r] + biasv[dt]) * maskr[r];
            c[dt][r] = f;
            feats_out[(pairBase + r + 8 * hi) * 128 + dt * 16 + nc] = f;
        }
    }

    // ---- Heads: preload this lane's Wt / Wc rows (d = dt*16 + nc) ----
    float wtv[8][8];
    float wcl[8];
#pragma unroll
    for (int dt = 0; dt < 8; ++dt) {
        const float4* p = (const float4*)(Wt + (size_t)(dt * 16 + nc) * 8);
        float4 lo = p[0], hh = p[1];
        wtv[dt][0] = lo.x; wtv[dt][1] = lo.y; wtv[dt][2] = lo.z; wtv[dt][3] = lo.w;
        wtv[dt][4] = hh.x; wtv[dt][5] = hh.y; wtv[dt][6] = hh.z; wtv[dt][7] = hh.w;
        wcl[dt] = Wc[dt * 16 + nc];
    }
    const float btv = bt[nc & 7];
    const float bcv = bc[0];

#pragma unroll
    for (int r = 0; r < 8; ++r) {
        const int m = r + 8 * hi;
        // causal head: dot over 128 d, reduced across 16-lane column group
        float pc = 0.0f;
#pragma unroll
        for (int dt = 0; dt < 8; ++dt) pc = fmaf(c[dt][r], wcl[dt], pc);
        pc += __shfl_xor(pc, 1, 32);
        pc += __shfl_xor(pc, 2, 32);
        pc += __shfl_xor(pc, 4, 32);
        pc += __shfl_xor(pc, 8, 32);
        float sg = 1.0f / (1.0f + expf(-(pc + bcv))) * maskr[r];
        if (nc == 0) causal_out[pairBase + m] = sg;

        // itypes head: 8 output channels
#pragma unroll
        for (int k = 0; k < 8; ++k) {
            float p = 0.0f;
#pragma unroll
            for (int dt = 0; dt < 8; ++dt) p = fmaf(c[dt][r], wtv[dt][k], p);
            p += __shfl_xor(p, 1, 32);
            p += __shfl_xor(p, 2, 32);
            p += __shfl_xor(p, 4, 32);
            p += __shfl_xor(p, 8, 32);
            if (nc == k) itypes_out[(pairBase + m) * 8 + k] = p + btv;
        }
    }
}

// ---------------------------------------------------------------------------
extern "C" void kernel_launch(void* const* d_in, const int* in_sizes, int n_in,
                              void* d_out, int out_size, void* d_ws, size_t ws_size,
                              hipStream_t stream) {
    const float* slots     = (const float*)d_in[0];
    const float* positions = (const float*)d_in[1];
    const float* em        = (const float*)d_in[2];
    const float* W1        = (const float*)d_in[3];
    const float* b1        = (const float*)d_in[4];
    const float* W2        = (const float*)d_in[5];
    const float* b2        = (const float*)d_in[6];
    const float* Wt        = (const float*)d_in[7];
    const float* bt        = (const float*)d_in[8];
    const float* Wc        = (const float*)d_in[9];
    const float* bc        = (const float*)d_in[10];

    float* out    = (float*)d_out;
    float* feats  = out;                        // 32*64*64*128
    float* itypes = out + (size_t)16777216;     // 32*64*64*8
    float* causal = out + (size_t)17825792;     // 32*64*64

    float* pa = (float*)d_ws;                   // [2048,256]
    float* pb = pa + (size_t)2048 * 256;        // [2048,256]

    // Kernel 1: PA/PB precompute (one wave per 16x16 tile)
    ie_k1_precompute<<<dim3(128, 16, 2), 32, 0, stream>>>(slots, W1, b1, pa);

    // Kernel 2: fused pairwise MLP + heads
    ie_k2_main<<<dim3(32, 4, 8), 256, 0, stream>>>(
        positions, em, W1, W2, b2, Wt, bt, Wc, bc, pa, pb,
        feats, itypes, causal);
}